// Projector_16879221473931
// MI455X (gfx1250) — compile-verified
//
#include <hip/hip_runtime.h>

// ---------------------------------------------------------------------------
// MI455X (gfx1250) fused cross-attention projector, fp32 WMMA (16x16x4).
// Wave32 fragment layouts (ISA 7.12.2):
//   A (16x4):  lanes 0-15 -> M=lane, K={0,1}; lanes 16-31 -> M=lane-16, K={2,3}
//   B (4x16):  lanes 0-15 -> N=lane, K={0,1}; lanes 16-31 -> N=lane-16, K={2,3}
//   C/D (16x16): vgpr g -> M = g (+8 for lanes 16-31), N = lane&15
// All LDS operand tiles are laid out so each fragment's (k, k+1) pair is
// contiguous + 8B aligned -> single ds_load_b64 per operand, no v_mov fixups.
// ---------------------------------------------------------------------------

typedef float v2f __attribute__((ext_vector_type(2)));
typedef float v8f __attribute__((ext_vector_type(8)));

#define WMMA_F32(a, b, c) \
  __builtin_amdgcn_wmma_f32_16x16x4_f32(false, (a), false, (b), (short)0, (c), false, false)

#define NB   32      // batch
#define CCH  512     // channels C == CT
#define LL   77      // tokens
#define LP   80      // padded L (multiple of 16)
#define TS   34      // even stride for transposed [l][c]/[l][d] tiles
#define SS82 82      // even stride for scores / V rows
#define HW   1024    // H*W
#define NH   16      // heads
#define DH   32      // head dim

// ===========================================================================
// Kernel 1: K/V projection.  out[n,o,l] = sum_c W[o,c]*tok[n,c,l] + b[o]
// tok is token (N,L,CT) reinterpreted flat as (N,CT,L) (reference reshape).
// Grid: (4 row-blocks of 128, N, {K,V}).  Block: 256 thr = 8 waves.
// Wave w owns a 16-row M-tile x 80 padded cols (5 N-tiles).
// Token chunk double-buffered in LDS, TRANSPOSED [l][c] so B fragments are
// single b64 loads.  W rows (A) come straight from global (L2-hot, b64).
// ===========================================================================
__global__ __launch_bounds__(256) void kv_gemm_kernel(
    const float* __restrict__ token,
    const float* __restrict__ Wk, const float* __restrict__ bk,
    const float* __restrict__ Wv, const float* __restrict__ bv,
    float* __restrict__ kbuf, float* __restrict__ vbuf)
{
  const int mblk = blockIdx.x;            // 0..3
  const int n    = blockIdx.y;
  const int mat  = blockIdx.z;            // 0 = K, 1 = V
  const float* W    = mat ? Wv : Wk;
  const float* bias = mat ? bv : bk;
  float* outb = (mat ? vbuf : kbuf) + n * (CCH * LP);

  __shared__ float sTokT[2][LP * TS];     // [l][c], c-chunk of 32

  const int tid   = threadIdx.x;
  const int lane  = tid & 31;
  const int wave  = tid >> 5;             // 0..7 -> M-tile
  const int ln    = lane & 15;
  const int kh    = (lane >> 4) << 1;     // K offset 0 or 2
  const int orow0 = mblk * 128 + wave * 16;

  v8f acc[5] = {};
  const float* tokn = token + n * (CCH * LL);

  // preload chunk 0 (c = 0..31); transpose-store, zero-pad l = 77..79
  for (int idx = tid; idx < 32 * LP; idx += 256) {
    int cc = idx / LP, l = idx - cc * LP;
    sTokT[0][l * TS + cc] = (l < LL) ? tokn[cc * LL + l] : 0.f;
  }
  __syncthreads();

  for (int ch = 0; ch < 16; ++ch) {
    const int cur = ch & 1;
    if (ch < 15) {                         // prefetch next chunk
      const int c0n = (ch + 1) * 32;
      for (int idx = tid; idx < 32 * LP; idx += 256) {
        int cc = idx / LP, l = idx - cc * LP;
        sTokT[cur ^ 1][l * TS + cc] = (l < LL) ? tokn[(c0n + cc) * LL + l] : 0.f;
      }
    }
    const int c0 = ch * 32;
    const float* sT = sTokT[cur];
    for (int kk = 0; kk < 32; kk += 4) {
      v2f a = *(const v2f*)(W + (orow0 + ln) * CCH + c0 + kk + kh);   // b64
#pragma unroll
      for (int t = 0; t < 5; ++t) {
        v2f b = *(const v2f*)(sT + (t * 16 + ln) * TS + kk + kh);     // b64
        acc[t] = WMMA_F32(a, b, acc[t]);
      }
    }
    __syncthreads();
  }

  // bias + store (row stride LP=80); lanes 0-15 write 64B-coalesced runs
#pragma unroll
  for (int t = 0; t < 5; ++t) {
#pragma unroll
    for (int g = 0; g < 8; ++g) {
      const int m = g + ((lane >> 4) << 3);
      const int o = orow0 + m;
      outb[o * LP + t * 16 + ln] = acc[t][g] + bias[o];
    }
  }
}

// ===========================================================================
// Kernel 2: fused Q-projection + attention + residual.
// Grid: (8 pixel tiles of 128, N, 16 heads).  Block: 256 thr = 8 waves.
//   stage1: q(128x32) = F^T(128x512) @ Wq_h^T(512x32)   (WMMA, K=512)
//   stage2: scores(128x80) = q @ K_h(32x80)             (WMMA, K=32)
//   softmax over L=77 (one thread per row, stride-82 conflict-free)
//   stage4: proj(128x32) = coef(128x80) @ V_h^T(80x32)  (WMMA, K=80)
//   epilogue: out = feature + proj  (coalesced NCHW stores via LDS)
// Feature is re-read per head but stays L2-resident (192 MB L2 >> 64 MB).
// ===========================================================================
__global__ __launch_bounds__(256) void attn_kernel(
    const float* __restrict__ feat,
    const float* __restrict__ Wq, const float* __restrict__ bq,
    const float* __restrict__ kbuf, const float* __restrict__ vbuf,
    float* __restrict__ out)
{
  const int pt = blockIdx.x;              // pixel tile (128 pixels)
  const int n  = blockIdx.y;
  const int h  = blockIdx.z;

  __shared__ float sWq[32 * 516];         // Wq head slice [d][c], stride 516
  __shared__ float sQ [128 * 36];         // q tile [p][d] (reused for proj, stride 33)
  __shared__ float sS [128 * SS82];       // scores [p][l], even stride 82
  __shared__ float sKT[LP * TS];          // K head tile TRANSPOSED [l][d]
  __shared__ float sV [32 * SS82];        // V head tile [d][l], even stride 82

  const int tid  = threadIdx.x;
  const int lane = tid & 31;
  const int wave = tid >> 5;
  const int ln   = lane & 15;
  const int kh   = (lane >> 4) << 1;
  const int hi8  = (lane >> 4) << 3;

  const float scale = 0.17677669529663687f;   // 1/sqrt(32)
  const float* fb = feat + n * (CCH * HW) + pt * 128;

  // ---- stage weights (pre-scaled) and K/V head tiles into LDS
  for (int idx = tid; idx < 32 * CCH; idx += 256) {
    int d = idx >> 9, c = idx & 511;
    sWq[d * 516 + c] = Wq[(h * DH + d) * CCH + c] * scale;
  }
  for (int idx = tid; idx < 32 * LP; idx += 256) {
    int d = idx / LP, l = idx - d * LP;
    sKT[l * TS + d]   = kbuf[n * (CCH * LP) + (h * DH + d) * LP + l];
    sV[d * SS82 + l]  = vbuf[n * (CCH * LP) + (h * DH + d) * LP + l];
  }
  __syncthreads();

  // ---- stage 1: q tile; wave w owns pixels [16w,16w+16), both d N-tiles
  {
    v8f q0 = {}, q1 = {};
    const int p0 = wave * 16;
    for (int c0 = 0; c0 < CCH; c0 += 4) {
      v2f a;                                        // A[m=p][k=c] = F[c][p]
      const float* ap = fb + (c0 + kh) * HW + p0 + ln;
      a.x = ap[0]; a.y = ap[HW];
      v2f b0 = *(const v2f*)(sWq + ln * 516 + c0 + kh);          // b64
      v2f b1 = *(const v2f*)(sWq + (16 + ln) * 516 + c0 + kh);   // b64
      q0 = WMMA_F32(a, b0, q0);
      q1 = WMMA_F32(a, b1, q1);
    }
    const float bq0 = bq[h * DH + ln] * scale;
    const float bq1 = bq[h * DH + 16 + ln] * scale;
#pragma unroll
    for (int g = 0; g < 8; ++g) {
      const int m = p0 + g + hi8;
      sQ[m * 36 + ln]      = q0[g] + bq0;
      sQ[m * 36 + 16 + ln] = q1[g] + bq1;
    }
  }
  __syncthreads();

  // ---- stage 2: scores = q @ K_h
  {
    const int p0 = wave * 16;
    v8f s[5] = {};
    for (int k0 = 0; k0 < DH; k0 += 4) {
      v2f a = *(const v2f*)(sQ + (p0 + ln) * 36 + k0 + kh);      // A[m=p][k=d]
#pragma unroll
      for (int t = 0; t < 5; ++t) {
        v2f b = *(const v2f*)(sKT + (t * 16 + ln) * TS + k0 + kh); // B[k=d][n=l]
        s[t] = WMMA_F32(a, b, s[t]);
      }
    }
#pragma unroll
    for (int t = 0; t < 5; ++t)
#pragma unroll
      for (int g = 0; g < 8; ++g)
        sS[(p0 + g + hi8) * SS82 + t * 16 + ln] = s[t][g];
  }
  __syncthreads();

  // ---- softmax over L=77 (stride 82 -> 18p mod 64 is a lane permutation)
  if (tid < 128) {
    float* row = sS + tid * SS82;
    float mx = row[0];
    for (int l = 1; l < LL; ++l) mx = fmaxf(mx, row[l]);
    float sum = 0.f;
    for (int l = 0; l < LL; ++l) { float e = __expf(row[l] - mx); row[l] = e; sum += e; }
    const float inv = 1.f / sum;
    for (int l = 0; l < LL; ++l) row[l] *= inv;
    row[77] = 0.f; row[78] = 0.f; row[79] = 0.f;    // kill pad columns
  }
  __syncthreads();

  // ---- stage 4: proj = coef @ V_h^T  (K = 80, pad cols contribute 0)
  {
    const int p0 = wave * 16;
    v8f o0 = {}, o1 = {};
    for (int l0 = 0; l0 < LP; l0 += 4) {
      v2f a  = *(const v2f*)(sS + (p0 + ln) * SS82 + l0 + kh);   // A[m=p][k=l]
      v2f b0 = *(const v2f*)(sV + ln * SS82 + l0 + kh);          // B[k=l][n=d]
      v2f b1 = *(const v2f*)(sV + (16 + ln) * SS82 + l0 + kh);
      o0 = WMMA_F32(a, b0, o0);
      o1 = WMMA_F32(a, b1, o1);
    }
    float* sP = sQ;                                 // reuse q buffer, stride 33
#pragma unroll
    for (int g = 0; g < 8; ++g) {
      const int m = p0 + g + hi8;
      sP[m * 33 + ln]      = o0[g];
      sP[m * 33 + 16 + ln] = o1[g];
    }
  }
  __syncthreads();

  // ---- epilogue: out[n, h*32+d, pt*128+p] = feature + proj  (coalesced)
  {
    const float* fres = fb + (h * DH) * HW;
    float* ob = out + n * (CCH * HW) + (h * DH) * HW + pt * 128;
    for (int idx = tid; idx < DH * 128; idx += 256) {
      const int d = idx >> 7, p = idx & 127;
      ob[d * HW + p] = fres[d * HW + p] + sQ[p * 33 + d];
    }
  }
}

// ===========================================================================
extern "C" void kernel_launch(void* const* d_in, const int* in_sizes, int n_in,
                              void* d_out, int out_size, void* d_ws, size_t ws_size,
                              hipStream_t stream) {
  const float* feature = (const float*)d_in[0];
  const float* token   = (const float*)d_in[1];
  const float* Wv      = (const float*)d_in[2];
  const float* bv      = (const float*)d_in[3];
  const float* Wk      = (const float*)d_in[4];
  const float* bk      = (const float*)d_in[5];
  const float* Wq      = (const float*)d_in[6];
  const float* bq      = (const float*)d_in[7];
  float* out = (float*)d_out;

  float* kbuf = (float*)d_ws;                       // 32*512*80 f32 = 5.24 MB
  float* vbuf = kbuf + (size_t)NB * CCH * LP;       // another 5.24 MB

  dim3 gkv(4, NB, 2);
  kv_gemm_kernel<<<gkv, 256, 0, stream>>>(token, Wk, bk, Wv, bv, kbuf, vbuf);

  dim3 gat(8, NB, NH);
  attn_kernel<<<gat, 256, 0, stream>>>(feature, Wq, bq, kbuf, vbuf, out);

  // tuple output: (out, token) -> copy token passthrough after the big tensor
  hipMemcpyAsync(out + (size_t)NB * CCH * HW, token,
                 (size_t)NB * LL * CCH * sizeof(float),
                 hipMemcpyDeviceToDevice, stream);
}